// EnhancedCircuitLoss_52621939310984
// MI455X (gfx1250) — compile-verified
//
#include <hip/hip_runtime.h>
#include <hip/hip_bf16.h>
#include <math.h>

// Problem constants (from reference): B=256, L=1024, D=256.
#define B_ 256
#define L_ 1024
#define D_ 256

typedef float v8f __attribute__((ext_vector_type(8)));
typedef float v2f __attribute__((ext_vector_type(2)));

__device__ __forceinline__ float readlane_f(float v, int lane) {
  return __int_as_float(__builtin_amdgcn_readlane(__float_as_int(v), lane));
}

// ---------------------------------------------------------------------------
// CDNA5 wave32 reduction using V_WMMA_F32_16X16X4_F32:
//   A(16x4) holds the 32 per-lane partials (K-slot 1 zeroed), B(4x16) = ones
//   =>  D[i][j] = rowsum_i(A).  Total = sum of column 0, which lives on
//   lane 0 (rows 0-7, VGPR 0-7) and lane 16 (rows 8-15).  fp32-exact.
//   EXEC must be all ones at call sites (calls are unconditional).
// ---------------------------------------------------------------------------
__device__ __forceinline__ float wave_reduce_sum(float v) {
  v2f a; a[0] = v;    a[1] = 0.0f;
  v2f o; o[0] = 1.0f; o[1] = 1.0f;
  v8f c = {0.f, 0.f, 0.f, 0.f, 0.f, 0.f, 0.f, 0.f};
  c = __builtin_amdgcn_wmma_f32_16x16x4_f32(false, a, false, o, (short)0, c,
                                            false, false);
  float s = c[0] + c[1] + c[2] + c[3] + c[4] + c[5] + c[6] + c[7];
  return readlane_f(s, 0) + readlane_f(s, 16);
}

template <int NW>
__device__ __forceinline__ float block_reduce_sum(float v, float* buf) {
  float w = wave_reduce_sum(v);
  const int lane = threadIdx.x & 31;
  const int wid  = threadIdx.x >> 5;
  __syncthreads();                 // protect buf reuse across calls
  if (lane == 0) buf[wid] = w;
  __syncthreads();
  float s = 0.0f;
  if (threadIdx.x == 0) {
#pragma unroll
    for (int i = 0; i < NW; ++i) s += buf[i];
  }
  return s;                        // valid on thread 0
}

// ---------------------------------------------------------------------------
// CDNA5 async global->LDS copy (ASYNCcnt-tracked).  VDST = LDS byte address,
// VADDR = 64-bit global address, no SADDR.
// ---------------------------------------------------------------------------
__device__ __forceinline__ void async_load_b128_to_lds(const float* gaddr,
                                                       unsigned lds_byte_off) {
  unsigned long long ga = (unsigned long long)(const void*)gaddr;
  asm volatile("global_load_async_to_lds_b128 %0, %1, off"
               :: "v"(lds_byte_off), "v"(ga)
               : "memory");
}

__device__ __forceinline__ void wait_asynccnt_le3() {
  asm volatile("s_wait_asynccnt 0x3" ::: "memory");
}
__device__ __forceinline__ void wait_asynccnt_le2() {
  asm volatile("s_wait_asynccnt 0x2" ::: "memory");
}
__device__ __forceinline__ void wait_asynccnt_le1() {
  asm volatile("s_wait_asynccnt 0x1" ::: "memory");
}
__device__ __forceinline__ void wait_asynccnt_0() {
  asm volatile("s_wait_asynccnt 0x0" ::: "memory");
}
__device__ __forceinline__ void wait_dscnt_0() {
  asm volatile("s_wait_dscnt 0x0" ::: "memory");
}

// ---------------------------------------------------------------------------
// Main kernel: one block per (batch b, D-half dh).  512 threads = 16 waves.
// Streams E[b, :, dh*128 : dh*128+128] through a 4-deep async LDS pipeline of
// 16-row x 128-float tiles (8 KB each).  Tile rows are WAVE-PRIVATE: wave p
// both copies and consumes row p of every tile, so no barriers are needed in
// the steady-state loop — each wave is ordered by its own s_wait_asynccnt
// (RAW) and s_wait_dscnt (WAR on buffer reuse).
// Accumulates Q4 (for |Q|^2) and s2 = sum (c*e)^2.
// ---------------------------------------------------------------------------
__global__ __launch_bounds__(512)
void circuit_loss_main(const float* __restrict__ final_pred,
                       const float* __restrict__ step_preds,
                       const float* __restrict__ uncertainty,
                       const float* __restrict__ area_targets,
                       const float* __restrict__ emb,
                       float* __restrict__ ws) {
  __shared__ float  sp_s[L_];
  __shared__ float  at_s[L_];
  __shared__ float  c_s[L_];
  __shared__ float  tile_s[4][16 * 128];   // 4 x 8 KB pipeline buffers
  __shared__ float4 qcomb[16 * 32];        // per-phase Q partials
  __shared__ float  red_s[16];

  const int tid = threadIdx.x;
  const int b   = blockIdx.x;
  const int dh  = blockIdx.y;              // which 128-wide half of D

  const size_t ebase = ((size_t)b * L_) * D_ + (size_t)dh * 128;
  const int p = tid >> 5;                  // row within 16-row tile (== wave id)
  const int g = tid & 31;                  // float4 group within half-row

  unsigned tb[4];
#pragma unroll
  for (int k = 0; k < 4; ++k)
    tb[k] = (unsigned)(unsigned long long)(const void*)&tile_s[k][0];

  const float* gsrc0 = emb + ebase + (size_t)p * D_ + g * 4;  // tile 0, row p

  // Prologue: fill the pipeline with tiles 0..2 (overlaps cheap-term work).
  async_load_b128_to_lds(gsrc0,              tb[0] + tid * 16);
  async_load_b128_to_lds(gsrc0 + 16 * D_,    tb[1] + tid * 16);
  async_load_b128_to_lds(gsrc0 + 32 * D_,    tb[2] + tid * 16);

  // ---- stage step_preds / area_targets, compute c = sq_err + cheap terms ----
  const float* spb = step_preds   + (size_t)b * L_;
  const float* atb = area_targets + (size_t)b * L_;
  sp_s[tid]       = spb[tid];
  sp_s[tid + 512] = spb[tid + 512];
  at_s[tid]       = atb[tid];
  at_s[tid + 512] = atb[tid + 512];
  __syncthreads();

  float csum = 0.f, rel = 0.f, crit = 0.f;
#pragma unroll
  for (int k = 0; k < 2; ++k) {
    const int l = tid + k * 512;
    const float s = sp_s[l], a = at_s[l];
    const float d = s - a;
    const float cc = d * d;
    c_s[l] = cc;
    csum += cc;
    if (l >= 1) {
      const float sm = sp_s[l - 1], am = at_s[l - 1];
      const float pr = (s - sm) / (sm + 1e-6f);
      const float tr = (a - am) / (am + 1e-6f);
      const float dd = pr - tr;
      rel = fmaf(dd, dd, rel);
      const float ae = fabsf(d);
      const float ch = fabsf(a - am);
      crit = fmaf(ae, 1.0f + ch, crit);
    }
  }

  float finv = 0.f, uncv = 0.f;
  if (tid == 0 && dh == 0) {
    const float fp = final_pred[b];
    const float ft = at_s[L_ - 1];
    const float u  = uncertainty[b];
    float fe = fp - ft; fe *= fe;
    finv = fe;
    uncv = logf(u) + fe / u;
  }
  __syncthreads();  // c_s ready for everyone (read-only afterwards)

  // ---- barrier-free pipelined stream over 64 tiles ----
  float4 Q = {0.f, 0.f, 0.f, 0.f};
  float  s2 = 0.f;
  const int g4 = g * 4;

  auto consume = [&](int T) {
    const float* tbuf = &tile_s[T & 3][0];
    const float  cl   = c_s[T * 16 + p];
    const float4 e    = *(const float4*)&tbuf[p * 128 + g4];
    const float u0 = cl * e.x, u1 = cl * e.y, u2 = cl * e.z, u3 = cl * e.w;
    Q.x += u0; Q.y += u1; Q.z += u2; Q.w += u3;
    s2 = fmaf(u0, u0, s2);
    s2 = fmaf(u1, u1, s2);
    s2 = fmaf(u2, u2, s2);
    s2 = fmaf(u3, u3, s2);
  };

  for (int T = 0; T < 61; ++T) {
    // Issue tile T+3 into buf[(T+3)&3] (same buffer tile T-1 used): make sure
    // our own ds_loads from it have drained first (WAR).
    wait_dscnt_0();
    async_load_b128_to_lds(gsrc0 + (size_t)(T + 3) * 16 * D_,
                           tb[(T + 3) & 3] + tid * 16);
    wait_asynccnt_le3();    // in-order completion => tile T has landed (RAW)
    consume(T);
  }
  wait_asynccnt_le2(); consume(61);
  wait_asynccnt_le1(); consume(62);
  wait_asynccnt_0();   consume(63);

  // ---- combine Q across the 16 row-phases, then |Q|^2 partial ----
  __syncthreads();
  qcomb[p * 32 + g] = Q;
  __syncthreads();
  float qsq = 0.f;
  if (tid < 32) {
    float4 qt = {0.f, 0.f, 0.f, 0.f};
#pragma unroll
    for (int pp = 0; pp < 16; ++pp) {
      const float4 qq = qcomb[pp * 32 + tid];
      qt.x += qq.x; qt.y += qq.y; qt.z += qq.z; qt.w += qq.w;
    }
    qsq = qt.x * qt.x + qt.y * qt.y + qt.z * qt.z + qt.w * qt.w;
  }

  const float q2t   = block_reduce_sum<16>(qsq,  red_s);
  const float s2t   = block_reduce_sum<16>(s2,   red_s);
  const float cst   = block_reduce_sum<16>(csum, red_s);
  const float relt  = block_reduce_sum<16>(rel,  red_s);
  const float critt = block_reduce_sum<16>(crit, red_s);

  if (tid == 0) {
    ws[       b * 2 + dh] = q2t;                // |Q|^2 partial per (b, dh)
    ws[512  + b * 2 + dh] = s2t;                // sum (c e)^2 partial
    if (dh == 0) {
      ws[1024 + b] = cst;                       // sum c         (step)
      ws[1280 + b] = relt;                      // rel partial
      ws[1536 + b] = critt;                     // critical partial
      ws[1792 + b] = finv;                      // (fp - ft)^2
      ws[2048 + b] = uncv;                      // log u + fe/u
    }
  }
}

// ---------------------------------------------------------------------------
// Finalize: combine 256 per-batch partials with the loss weights.
//   seq_dep = mean(c) + (|Q|^2 - s2)/2 / (B L);  0.2*seq_dep folds into
//   1.2*step_mean + 0.1*(q2 - s2)/(B L).  uncertainty carries 0.3*0.5 = 0.15.
// ---------------------------------------------------------------------------
__global__ __launch_bounds__(256)
void circuit_loss_finalize(const float* __restrict__ ws, float* __restrict__ out) {
  __shared__ float red_s[8];
  const int b = threadIdx.x;

  const float q2 = ws[2 * b] + ws[2 * b + 1];
  const float s2 = ws[512 + 2 * b] + ws[512 + 2 * b + 1];
  const float cs = ws[1024 + b];
  const float rl = ws[1280 + b];
  const float cr = ws[1536 + b];
  const float fn = ws[1792 + b];
  const float un = ws[2048 + b];

  const float invB   = 1.0f / 256.0f;
  const float invBL  = 1.0f / (256.0f * 1024.0f);
  const float invBL1 = 1.0f / (256.0f * 1023.0f);

  const float t = fn * invB
                + rl * invBL1
                + 1.2f * cs * invBL
                + 0.3f * cr * invBL1
                + 0.1f * (q2 - s2) * invBL
                + 0.15f * un * invB;

  const float tot = block_reduce_sum<8>(t, red_s);
  if (b == 0) out[0] = tot;
}

extern "C" void kernel_launch(void* const* d_in, const int* in_sizes, int n_in,
                              void* d_out, int out_size, void* d_ws, size_t ws_size,
                              hipStream_t stream) {
  (void)in_sizes; (void)n_in; (void)out_size; (void)ws_size;
  const float* final_pred   = (const float*)d_in[0];
  const float* step_preds   = (const float*)d_in[1];
  const float* uncertainty  = (const float*)d_in[2];
  const float* area_targets = (const float*)d_in[3];
  const float* emb          = (const float*)d_in[4];
  // d_in[5] (recipes, int64) is unused by the reference loss.
  float* ws  = (float*)d_ws;    // 2304 floats used (~9.2 KB), all pure stores
  float* out = (float*)d_out;

  dim3 grid(B_, 2);             // 512 blocks x 16 waves
  circuit_loss_main<<<grid, 512, 0, stream>>>(final_pred, step_preds, uncertainty,
                                              area_targets, emb, ws);
  circuit_loss_finalize<<<1, 256, 0, stream>>>(ws, out);
}